// NAMTuringRecurrent_5076651344118
// MI455X (gfx1250) — compile-verified
//
#include <hip/hip_runtime.h>
#include <math.h>

// ---------------------------------------------------------------------------
// NAM Turing recurrent model for gfx1250 (MI455X), fp32, WMMA 16x16x4_f32.
//
// Pipeline (all on `stream`, launches serialize):
//   1. init barrier counter
//   2. gates_x = X @ Wih + bih + bhh          (WMMA GEMM, async-LDS staging)
//   3. V      = X @ Wv + bv                   (WMMA GEMM)
//   4. Kraw   = X @ Wk + bk ; normalize heads (WMMA GEMM + wave-reduce)
//   5. LSTM persistent kernel (128 blocks, Whh slice TDM-staged into LDS,
//      device-wide barrier per step, WMMA for h @ Whh)
//   6. Qh = hidden @ Wq[:512] + bq            (WMMA GEMM)
//   7. Ah = hidden @ Wa[:512] + ba            (simple VALU kernel, N=40)
//   8. tape kernel: 1 workgroup per batch, tape+key tape resident in LDS
//   9. out = read_outs @ Wo + bo              (WMMA GEMM) -> d_out
//
// Workspace requirement: ~119 MB (see offsets in kernel_launch).
// ---------------------------------------------------------------------------

typedef __attribute__((ext_vector_type(2))) float v2f;
typedef __attribute__((ext_vector_type(8))) float v8f;
typedef int v4i __attribute__((ext_vector_type(4)));
typedef unsigned int u32x4 __attribute__((ext_vector_type(4)));
typedef int i32x4 __attribute__((ext_vector_type(4)));
typedef int i32x8 __attribute__((ext_vector_type(8)));
typedef __attribute__((address_space(1))) void GAS;  // global void
typedef __attribute__((address_space(3))) void LAS;  // LDS void
typedef __attribute__((address_space(1))) v4i GV4i;  // global int4 (async b128)
typedef __attribute__((address_space(3))) v4i LV4i;  // LDS int4   (async b128)

#if defined(__gfx1250__) && __has_builtin(__builtin_amdgcn_global_load_async_to_lds_b128)
#define USE_ASYNC_LDS 1
#else
#define USE_ASYNC_LDS 0
#endif

#if USE_ASYNC_LDS && __has_builtin(__builtin_amdgcn_s_wait_asynccnt)
#define WAIT_ASYNC(n) __builtin_amdgcn_s_wait_asynccnt(n)
#elif USE_ASYNC_LDS
#define WAIT_ASYNC(n) asm volatile("s_wait_asynccnt %0" ::"n"(n))
#else
#define WAIT_ASYNC(n)
#endif

#if defined(__gfx1250__) && __has_builtin(__builtin_amdgcn_tensor_load_to_lds) && \
    __has_builtin(__builtin_amdgcn_s_wait_tensorcnt)
#define USE_TDM 1
#else
#define USE_TDM 0
#endif

#define SEQ   256
#define BATCH 32
#define DIM   512
#define NT    4
#define HDIM  64
#define TAPL  32
#define MDIM  256   // NT*HDIM
#define GDIM  2048  // 4*DIM
#define EPS_  1e-12f

__device__ __forceinline__ float sig_(float x) { return 1.0f / (1.0f + expf(-x)); }

// --------------------------- device-wide barrier ---------------------------
__device__ __forceinline__ void grid_sync(unsigned* bar, unsigned nb) {
  __syncthreads();
  if (threadIdx.x == 0) {
    __threadfence();
    unsigned prev = __hip_atomic_fetch_add(bar, 1u, __ATOMIC_ACQ_REL,
                                           __HIP_MEMORY_SCOPE_AGENT);
    unsigned target = (prev / nb + 1u) * nb;
    while (__hip_atomic_load(bar, __ATOMIC_ACQUIRE,
                             __HIP_MEMORY_SCOPE_AGENT) < target) {
      __builtin_amdgcn_s_sleep(2);
    }
  }
  __syncthreads();
}

__global__ void init_kernel(unsigned* bar) {
  if (threadIdx.x == 0 && blockIdx.x == 0) *bar = 0u;
}

// --------------------------- generic WMMA GEMM -----------------------------
// C[M,N] = A[M,K] @ B[K,N] + bias0 + bias1.  Requirements: M%128==0,
// N%64==0, K%32==0 (true for every call site here).
// Block: 256 threads = 8 waves; block tile 128(M) x 64(N); wave tile 16x64.
// B panels are double-buffered in LDS; on gfx1250 the panel copy goes
// straight to LDS via GLOBAL_LOAD_ASYNC_TO_LDS_B128 (tracked on ASYNCcnt),
// overlapping the current panel's 32 WMMAs.
__global__ __launch_bounds__(256) void gemm_wmma_f32(
    const float* __restrict__ A, const float* __restrict__ B,
    const float* __restrict__ bias0, const float* __restrict__ bias1,
    float* __restrict__ C, int M, int N, int K) {
  __shared__ __align__(16) float Bs[2][32][80];  // pad 64->80: bank rotation

  const int tid  = threadIdx.x;
  const int wave = tid >> 5;
  const int lane = tid & 31;
  const int kSel = (lane >> 4) << 1;  // lanes 0-15 -> K+0/K+1, 16-31 -> K+2/K+3
  const int colS = lane & 15;
  const int mBase = blockIdx.y * 128;
  const int nBase = blockIdx.x * 64;
  const float* __restrict__ Arow = A + (size_t)(mBase + wave * 16 + colS) * K;

  const int srow = tid >> 3;        // 0..31  (staging row)
  const int scol = (tid & 7) * 8;   // 0..56  (staging col, 8 floats)

  auto stage = [&](int buf, int kb) {
    const float* src = B + (size_t)(kb + srow) * N + nBase + scol;
#if USE_ASYNC_LDS
    __builtin_amdgcn_global_load_async_to_lds_b128(
        (GV4i*)(GAS*)src, (LV4i*)(LAS*)&Bs[buf][srow][scol], 0, 0);
    __builtin_amdgcn_global_load_async_to_lds_b128(
        (GV4i*)(GAS*)(src + 4), (LV4i*)(LAS*)&Bs[buf][srow][scol + 4], 0, 0);
#else
    float4 p0 = *(const float4*)(src);
    float4 p1 = *(const float4*)(src + 4);
    *(float4*)&Bs[buf][srow][scol]     = p0;
    *(float4*)&Bs[buf][srow][scol + 4] = p1;
#endif
  };

  v8f acc0 = {}, acc1 = {}, acc2 = {}, acc3 = {};

  stage(0, 0);  // prologue: panel 0 in flight
  for (int kb = 0; kb < K; kb += 32) {
    const int cur = (kb >> 5) & 1;
    if (kb + 32 < K) {
      stage(cur ^ 1, kb + 32);  // overlap next panel copy with compute
      WAIT_ASYNC(2);            // wave's 2 older (current-panel) loads done
    } else {
      WAIT_ASYNC(0);
    }
    __syncthreads();
#pragma unroll
    for (int kk = 0; kk < 32; kk += 4) {
      const int k0 = kk + kSel;
      v2f a = *(const v2f*)(Arow + kb + k0);
      v2f b0, b1, b2, b3;
      b0.x = Bs[cur][k0][colS];          b0.y = Bs[cur][k0 + 1][colS];
      b1.x = Bs[cur][k0][16 + colS];     b1.y = Bs[cur][k0 + 1][16 + colS];
      b2.x = Bs[cur][k0][32 + colS];     b2.y = Bs[cur][k0 + 1][32 + colS];
      b3.x = Bs[cur][k0][48 + colS];     b3.y = Bs[cur][k0 + 1][48 + colS];
      acc0 = __builtin_amdgcn_wmma_f32_16x16x4_f32(false, a, false, b0, (short)0, acc0, false, false);
      acc1 = __builtin_amdgcn_wmma_f32_16x16x4_f32(false, a, false, b1, (short)0, acc1, false, false);
      acc2 = __builtin_amdgcn_wmma_f32_16x16x4_f32(false, a, false, b2, (short)0, acc2, false, false);
      acc3 = __builtin_amdgcn_wmma_f32_16x16x4_f32(false, a, false, b3, (short)0, acc3, false, false);
    }
    __syncthreads();
  }

  // Epilogue: C/D layout -> VGPR r holds (M = r + (lane>=16)*8, N = lane&15)
  const int rBase = mBase + wave * 16 + ((lane >> 4) << 3);
  v8f accs[4] = {acc0, acc1, acc2, acc3};
#pragma unroll
  for (int n = 0; n < 4; ++n) {
    const int col = nBase + n * 16 + colS;
    float bsum = 0.0f;
    if (bias0) bsum += bias0[col];
    if (bias1) bsum += bias1[col];
#pragma unroll
    for (int r = 0; r < 8; ++r)
      C[(size_t)(rBase + r) * N + col] = accs[n][r] + bsum;
  }
}

// --------------------------- key normalization -----------------------------
// One wave per (s,b,t) head of 64 elements: v /= max(||v||, EPS)
__global__ __launch_bounds__(32) void knorm_kernel(float* __restrict__ K) {
  float* p = K + (size_t)blockIdx.x * HDIM;
  const int lane = threadIdx.x;
  float a = p[lane], b = p[lane + 32];
  float ss = a * a + b * b;
#pragma unroll
  for (int off = 16; off; off >>= 1) ss += __shfl_xor(ss, off, 32);
  float inv = 1.0f / fmaxf(sqrtf(ss), EPS_);
  p[lane] = a * inv;
  p[lane + 32] = b * inv;
}

// --------------------------- persistent LSTM -------------------------------
// 128 blocks x 64 threads (2 waves). Block bj owns h columns [bj*4, bj*4+4).
// Whh slice (512 x 16 interleaved gate columns) staged to LDS once via one
// TDM tensor_load_to_lds (tile x=4 contiguous, y=4 @stride 512, z=512
// @stride 2048 -> packed x-fastest == WhhS[k][g*4+t]), resident for all 256
// steps; per step: G = gates_x[s] + h_{s-1} @ WhhS via WMMA, then gate math.
__global__ __launch_bounds__(64) void lstm_kernel(
    const float* __restrict__ gates_x,  // [S][B][2048]
    const float* __restrict__ Whh,      // [512][2048]
    float* __restrict__ hidden,         // [S][B][512]
    unsigned* __restrict__ bar) {
  __shared__ __align__(16) float WhhS[512][16];  // col = g*4 + t
  __shared__ float Gs[32][20];  // padded: +32 bank rotation for hi-lanes
  __shared__ float cS[32][4];

  const int tid = threadIdx.x;
  const int bj  = blockIdx.x;  // 0..127

#if USE_TDM
  if (tid < 32) {  // one wave issues the tensor DMA
    unsigned ldsOff = (unsigned)(unsigned long long)(LAS*)&WhhS[0][0];
    unsigned long long ga =
        (unsigned long long)(const void*)(Whh + (size_t)bj * 4);
    u32x4 g0;
    g0[0] = 1u;                                        // count=1, load, user
    g0[1] = ldsOff;                                    // lds_addr
    g0[2] = (unsigned)(ga & 0xFFFFFFFFull);            // global_addr[31:0]
    g0[3] = (unsigned)((ga >> 32) & 0x01FFFFFFull)     // global_addr[56:32]
            | 0x80000000u;                             // type=2 ("image")
    i32x8 g1;
    g1[0] = 0x20000;          // data_size = 4 bytes
    g1[1] = (4 << 16);        // tensor_dim0 = 4   (x = t)
    g1[2] = (4 << 16);        // tensor_dim1 = 4   (y = gate)
    g1[3] = (4 << 16);        // tile_dim0   = 4
    g1[4] = 4 | (512 << 16);  // tile_dim1 = 4, tile_dim2 = 512 (z = k)
    g1[5] = 512;              // tensor_dim0_stride = 512 elements
    g1[6] = (2048 << 16);     // tensor_dim1_stride = 2048 elements (lo16)
    g1[7] = 0;                //                      (hi bits)
    i32x4 g2;
    g2[0] = 512;              // tensor_dim2 = 512
    g2[1] = 0; g2[2] = 0; g2[3] = 0;
    i32x4 g3 = {0, 0, 0, 0};
#if __clang_major__ >= 23
    i32x8 g4 = {0, 0, 0, 0, 0, 0, 0, 0};
    __builtin_amdgcn_tensor_load_to_lds(g0, g1, g2, g3, g4, 0);
#else
    __builtin_amdgcn_tensor_load_to_lds(g0, g1, g2, g3, 0);
#endif
    __builtin_amdgcn_s_wait_tensorcnt(0);
  }
#else
  for (int idx = tid; idx < 512 * 16; idx += 64) {
    int k = idx >> 4, col = idx & 15;
    int g = col >> 2, t = col & 3;
    WhhS[k][col] = Whh[(size_t)k * GDIM + g * DIM + bj * 4 + t];
  }
#endif
  for (int idx = tid; idx < 128; idx += 64) cS[idx >> 2][idx & 3] = 0.0f;
  __syncthreads();

  const int wave = tid >> 5;
  const int lane = tid & 31;
  const int kSel = (lane >> 4) << 1;
  const int colS = lane & 15;

  for (int s = 0; s < SEQ; ++s) {
    v8f acc = {};
    if (s > 0) {
      const float* __restrict__ arow =
          hidden + (size_t)(s - 1) * BATCH * DIM + (size_t)(wave * 16 + colS) * DIM;
#pragma unroll 8
      for (int k4 = 0; k4 < DIM; k4 += 4) {
        const int k0 = k4 + kSel;
        v2f a = *(const v2f*)(arow + k0);
        v2f b;
        b.x = WhhS[k0][colS];
        b.y = WhhS[k0 + 1][colS];
        acc = __builtin_amdgcn_wmma_f32_16x16x4_f32(false, a, false, b, (short)0, acc, false, false);
      }
    }
    const int rBase = wave * 16 + ((lane >> 4) << 3);
#pragma unroll
    for (int r = 0; r < 8; ++r) Gs[rBase + r][colS] = acc[r];
    __syncthreads();

    const float* __restrict__ gx = gates_x + (size_t)s * BATCH * GDIM;
    for (int e = tid; e < 128; e += 64) {
      int b = e >> 2, d = e & 3;
      int hc = bj * 4 + d;
      float gi = Gs[b][0 + d]  + gx[(size_t)b * GDIM + 0 * DIM + hc];
      float gf = Gs[b][4 + d]  + gx[(size_t)b * GDIM + 1 * DIM + hc];
      float gg = Gs[b][8 + d]  + gx[(size_t)b * GDIM + 2 * DIM + hc];
      float go = Gs[b][12 + d] + gx[(size_t)b * GDIM + 3 * DIM + hc];
      float c = sig_(gf) * cS[b][d] + sig_(gi) * tanhf(gg);
      cS[b][d] = c;
      hidden[(size_t)s * BATCH * DIM + (size_t)b * DIM + hc] = sig_(go) * tanhf(c);
    }
    grid_sync(bar, 128);  // hidden[s] complete device-wide before step s+1
  }
}

// --------------------------- Ah = h @ Wa[:512] + ba ------------------------
__global__ __launch_bounds__(256) void ah_kernel(
    const float* __restrict__ hidden, const float* __restrict__ Wa,
    const float* __restrict__ ba, float* __restrict__ Ah) {
  int idx = blockIdx.x * 256 + threadIdx.x;
  if (idx >= SEQ * BATCH * 40) return;
  int row = idx / 40, j = idx % 40;
  const float* h = hidden + (size_t)row * DIM;
  float acc = ba[j];
#pragma unroll 8
  for (int k = 0; k < DIM; ++k) acc += h[k] * Wa[(size_t)k * 40 + j];
  Ah[idx] = acc;
}

// --------------------------- tape recurrence -------------------------------
// One workgroup per batch element; tape + key tape fully LDS-resident.
__device__ __forceinline__ void softmax4_(const float* a, float* o) {
  float m = fmaxf(fmaxf(a[0], a[1]), fmaxf(a[2], a[3]));
  float e0 = expf(a[0] - m), e1 = expf(a[1] - m);
  float e2 = expf(a[2] - m), e3 = expf(a[3] - m);
  float inv = 1.0f / (e0 + e1 + e2 + e3);
  o[0] = e0 * inv; o[1] = e1 * inv; o[2] = e2 * inv; o[3] = e3 * inv;
}

__global__ __launch_bounds__(256) void tape_kernel(
    const float* __restrict__ Qh,   // [S][B][256]
    const float* __restrict__ Ah,   // [S][B][40]
    const float* __restrict__ Vb,   // [S][B][256]
    const float* __restrict__ Kb,   // [S][B][256] (normalized keys)
    const float* __restrict__ Wq,   // [768][256]  (rows 512.. used)
    const float* __restrict__ Wa,   // [768][40]   (rows 512.. used)
    float* __restrict__ rdouts,     // [S][B][256]
    float* __restrict__ out_tape) { // [L][B][256]
  __shared__ float tapeS[TAPL][MDIM];  // 32 KB
  __shared__ float keyS[TAPL][MDIM];   // 32 KB
  __shared__ float rposS[TAPL][NT];
  __shared__ float wposS[TAPL][NT];
  __shared__ float roS[MDIM];
  __shared__ float roNewS[MDIM];
  __shared__ float qS[MDIM];
  __shared__ float actS[NT][10];
  __shared__ float rwS[NT][2];
  __shared__ float rdirS[NT][4];
  __shared__ float wdirS[NT][4];
  __shared__ float jposS[TAPL][NT];
  __shared__ float jnInvS[NT];

  const int tid = threadIdx.x;  // 0..255
  const int b   = blockIdx.x;   // batch element

  for (int idx = tid; idx < TAPL * MDIM; idx += 256) {
    tapeS[idx >> 8][idx & 255] = 0.0f;
    keyS[idx >> 8][idx & 255] = 0.0f;
  }
  roS[tid] = 0.0f;
  if (tid < TAPL * NT) {
    int l = tid >> 2, t = tid & 3;
    float v = (l == 0) ? 1.0f : 0.0f;
    rposS[l][t] = v;
    wposS[l][t] = v;
  }
  __syncthreads();

  const int j = tid;        // 0..255 = t*64 + c
  const int tj = j >> 6;    // head of this thread's column

  for (int s = 0; s < SEQ; ++s) {
    const size_t sb = (size_t)s * BATCH + b;
    const float* Qh_s = Qh + sb * MDIM;
    const float* Ah_s = Ah + sb * 40;
    const float* V_s  = Vb + sb * MDIM;
    const float* K_s  = Kb + sb * MDIM;

    // Phase A: query/action = h-part (precomputed) + read_out @ W[512:]
    {
      float acc = Qh_s[j];
#pragma unroll 8
      for (int i = 0; i < MDIM; ++i)
        acc += roS[i] * Wq[(size_t)(512 + i) * MDIM + j];
      qS[j] = acc;
    }
    if (j < 40) {
      float acc = Ah_s[j];
#pragma unroll 8
      for (int i = 0; i < MDIM; ++i)
        acc += roS[i] * Wa[(size_t)(512 + i) * 40 + j];
      actS[j / 10][j % 10] = acc;
    }
    __syncthreads();

    // Phase B: gate nonlinearities
    if (tid < NT) {
      int t = tid;
      rwS[t][0] = sig_(actS[t][8]);
      rwS[t][1] = sig_(actS[t][9]);
      softmax4_(&actS[t][0], &rdirS[t][0]);
      softmax4_(&actS[t][4], &wdirS[t][0]);
    }
    __syncthreads();

    // Phase C: tape + key-tape write, read_out (old rpos) from updated tape
    {
      float ov = 0.0f, ok = 0.0f;
#pragma unroll 4
      for (int l = 0; l < TAPL; ++l) {
        float wp = wposS[l][tj];
        ov += tapeS[l][j] * wp;
        ok += keyS[l][j] * wp;
      }
      float w1 = rwS[tj][1];
      float dv = (V_s[j] - ov) * w1;
      float dk = (K_s[j] - ok) * w1;
      float rd = 0.0f;
#pragma unroll 4
      for (int l = 0; l < TAPL; ++l) {
        float wp = wposS[l][tj];
        float tv = tapeS[l][j] + wp * dv;
        tapeS[l][j] = tv;
        keyS[l][j] = keyS[l][j] + wp * dk;
        rd += tv * rposS[l][tj];
      }
      roNewS[j] = rd;
    }
    __syncthreads();

    // Phase D: jpos[l][t] = <key[l][t][:], query[t][:]>, then L-normalize
    if (tid < TAPL * NT) {
      int l = tid >> 2, t = tid & 3;
      float acc = 0.0f;
#pragma unroll 8
      for (int cc = 0; cc < HDIM; ++cc) {
        int c = (cc + l) & 63;  // rotate to avoid LDS bank conflicts
        acc += keyS[l][t * HDIM + c] * qS[t * HDIM + c];
      }
      jposS[l][t] = acc;
    }
    __syncthreads();
    if (tid < NT) {
      float ss = 0.0f;
#pragma unroll 4
      for (int l = 0; l < TAPL; ++l) {
        float x = jposS[l][tid];
        ss += x * x;
      }
      jnInvS[tid] = 1.0f / fmaxf(sqrtf(ss), EPS_);
    }
    __syncthreads();

    // Phase E: head-position updates (reads old rpos/wpos, then writes)
    float nrp = 0.0f, nwp = 0.0f;
    int pl = tid >> 2, pt = tid & 3;
    if (tid < TAPL * NT) {
      float jn  = jposS[pl][pt] * jnInvS[pt];
      float rp  = rposS[pl][pt];
      float prv = rposS[(pl + 1) & 31][pt];      // roll(rpos,-1)[l]
      float nxt = rposS[(pl + 31) & 31][pt];     // roll(rpos,+1)[l]
      float wp  = wposS[pl][pt];
      nwp = prv * wdirS[pt][0] + wp * wdirS[pt][1] + nxt * wdirS[pt][2] + jn * wdirS[pt][3];
      nrp = prv * rdirS[pt][0] + rp * rdirS[pt][1] + nxt * rdirS[pt][2] + jn * rdirS[pt][3];
    }
    __syncthreads();
    if (tid < TAPL * NT) {
      rposS[pl][pt] = nrp;
      wposS[pl][pt] = nwp;
    }

    // Phase F: scale and emit read_out
    {
      float ro = roNewS[j] * rwS[tj][0];
      roS[j] = ro;
      rdouts[sb * MDIM + j] = ro;
    }
    __syncthreads();
  }

  // Final tape dump: out_tape[l][b][m]
  for (int idx = tid; idx < TAPL * MDIM; idx += 256) {
    int l = idx >> 8, m = idx & 255;
    out_tape[((size_t)l * BATCH + b) * MDIM + m] = tapeS[l][m];
  }
}

// --------------------------- host launcher ---------------------------------
extern "C" void kernel_launch(void* const* d_in, const int* in_sizes, int n_in,
                              void* d_out, int out_size, void* d_ws, size_t ws_size,
                              hipStream_t stream) {
  const float* inputs = (const float*)d_in[0];
  const float* Wih = (const float*)d_in[1];
  const float* Whh = (const float*)d_in[2];
  const float* bih = (const float*)d_in[3];
  const float* bhh = (const float*)d_in[4];
  const float* Wa  = (const float*)d_in[5];
  const float* ba  = (const float*)d_in[6];
  const float* Wv  = (const float*)d_in[7];
  const float* bv  = (const float*)d_in[8];
  const float* Wk  = (const float*)d_in[9];
  const float* bk  = (const float*)d_in[10];
  const float* Wq  = (const float*)d_in[11];
  const float* bq  = (const float*)d_in[12];
  const float* Wo  = (const float*)d_in[13];
  const float* bo  = (const float*)d_in[14];
  float* out = (float*)d_out;

  // Workspace layout (floats). Total ~29.7M floats (~119 MB).
  float* ws = (float*)d_ws;
  size_t off = 0;
  float* gates_x = ws + off; off += (size_t)SEQ * BATCH * GDIM;  // 16.8M
  float* Vbuf    = ws + off; off += (size_t)SEQ * BATCH * MDIM;  // 2.1M
  float* Kbuf    = ws + off; off += (size_t)SEQ * BATCH * MDIM;  // 2.1M
  float* hidden  = ws + off; off += (size_t)SEQ * BATCH * DIM;   // 4.2M
  float* Qh      = ws + off; off += (size_t)SEQ * BATCH * MDIM;  // 2.1M
  float* Ah      = ws + off; off += (size_t)SEQ * BATCH * 40;    // 0.33M
  float* rdouts  = ws + off; off += (size_t)SEQ * BATCH * MDIM;  // 2.1M
  unsigned* bar  = (unsigned*)(ws + off);

  const int MROWS = SEQ * BATCH;  // 8192

  init_kernel<<<1, 32, 0, stream>>>(bar);

  gemm_wmma_f32<<<dim3(GDIM / 64, MROWS / 128), 256, 0, stream>>>(
      inputs, Wih, bih, bhh, gates_x, MROWS, GDIM, DIM);
  gemm_wmma_f32<<<dim3(MDIM / 64, MROWS / 128), 256, 0, stream>>>(
      inputs, Wv, bv, nullptr, Vbuf, MROWS, MDIM, DIM);
  gemm_wmma_f32<<<dim3(MDIM / 64, MROWS / 128), 256, 0, stream>>>(
      inputs, Wk, bk, nullptr, Kbuf, MROWS, MDIM, DIM);
  knorm_kernel<<<MROWS * NT, 32, 0, stream>>>(Kbuf);

  lstm_kernel<<<128, 64, 0, stream>>>(gates_x, Whh, hidden, bar);

  gemm_wmma_f32<<<dim3(MDIM / 64, MROWS / 128), 256, 0, stream>>>(
      hidden, Wq, bq, nullptr, Qh, MROWS, MDIM, DIM);
  ah_kernel<<<(MROWS * 40 + 255) / 256, 256, 0, stream>>>(hidden, Wa, ba, Ah);

  tape_kernel<<<BATCH, 256, 0, stream>>>(
      Qh, Ah, Vbuf, Kbuf, Wq, Wa, rdouts, out + (size_t)MROWS * DIM);

  gemm_wmma_f32<<<dim3(DIM / 64, MROWS / 128), 256, 0, stream>>>(
      rdouts, Wo, bo, nullptr, out, MROWS, DIM, MDIM);
}